// TotalLoss_12756052869800
// MI455X (gfx1250) — compile-verified
//
#include <hip/hip_runtime.h>
#include <hip/hip_bf16.h>
#include <math.h>

#define B_ 256
#define K_ 32
#define C_ 2048
#define NCLS_ 20
#define MARGIN_ 150.0f
#define TRIP_MARGIN_ 0.8f
#define T_ 0.07f

#define RPC 2                 // rows per TDM chunk
#define NCHUNK (K_ / RPC)     // 16

typedef __attribute__((ext_vector_type(2))) float v2f;
typedef __attribute__((ext_vector_type(8))) float v8f;
typedef __attribute__((ext_vector_type(4))) unsigned u32x4;
typedef __attribute__((ext_vector_type(8))) int i32x8;
typedef __attribute__((ext_vector_type(4))) int i32x4;

// ---------- TDM: DMA 'nrows' consecutive rows of C_ floats from global to LDS ----------
__device__ __forceinline__ void tdm_load_rows(const float* gsrc, void* lds_dst, int nrows) {
    unsigned long long ga = (unsigned long long)(uintptr_t)gsrc;
    unsigned la = (unsigned)(uintptr_t)lds_dst;   // low 32 bits of generic LDS ptr = LDS offset
    u32x4 g0;
    g0[0] = 1u;                                          // count=1, is_restore=0, gather=0
    g0[1] = la;                                          // lds_addr (bytes)
    g0[2] = (unsigned)ga;                                // global_addr[31:0]
    g0[3] = (unsigned)((ga >> 32) & 0x01FFFFFFu) | (2u << 30);  // global_addr[56:32] | type=2
    i32x8 g1;
    g1[0] = 0x20000;             // workgroup_mask=0, data_size=2 (4B), no barrier/iter/pad
    g1[1] = (int)(C_ << 16);     // [63:48] tensor_dim0[15:0] = 2048
    g1[2] = (int)(K_ << 16);     // [95:80] tensor_dim1[15:0] = 32
    g1[3] = (int)(C_ << 16);     // [127:112] tile_dim0 = 2048
    g1[4] = nrows;               // [143:128] tile_dim1
    g1[5] = C_;                  // [191:160] tensor_dim0_stride = 2048
    g1[6] = 0;
    g1[7] = 0;
    i32x4 z4 = {0, 0, 0, 0};
#if defined(__clang_major__) && (__clang_major__ >= 23)
    i32x8 z8 = {0, 0, 0, 0, 0, 0, 0, 0};
    __builtin_amdgcn_tensor_load_to_lds(g0, g1, z4, z4, z8, 0);
#else
    __builtin_amdgcn_tensor_load_to_lds(g0, g1, z4, z4, 0);
#endif
}

// ---------- Pass: mean over K (axis=1) for a [B,K,C] tensor (NA/NB only need this) ----------
__global__ void mean_over_k(const float* __restrict__ X, float* __restrict__ out) {
    int gid = blockIdx.x * blockDim.x + threadIdx.x;   // over B*C
    int b = gid >> 11;                                  // / C_ (2048)
    int c = gid & (C_ - 1);
    const float* p = X + (size_t)b * K_ * C_ + c;
    float s = 0.f;
#pragma unroll
    for (int k = 0; k < K_; ++k) s += p[(size_t)k * C_];
    out[gid] = s * (1.0f / K_);
}

// ---------- Fused single-read pass over a [B,K,C] tensor (one block per b), TDM-staged ----------
// Produces: meanX[b,c] (mean over k), sq[b,k]=sum_c x^2, dot[b,k]=sum_c q[b,c]*x, act[b,k]=mean_c x
__global__ void __launch_bounds__(256) fused_stats(const float* __restrict__ X,
                                                   const float* __restrict__ q,
                                                   float* __restrict__ meanX,
                                                   float* __restrict__ sq,
                                                   float* __restrict__ dot,
                                                   float* __restrict__ act) {
    __shared__ float lbuf[2][RPC * C_];   // 2 x 16 KB double buffer
    __shared__ float wred[6][8];
    int b = blockIdx.x, t = threadIdx.x;
    int wave = t >> 5, lane = t & 31;
    const float* xbase = X + (size_t)b * K_ * C_;

    float qreg[8], colsum[8];
#pragma unroll
    for (int j = 0; j < 8; ++j) {
        qreg[j] = q[(size_t)b * C_ + t + 256 * j];
        colsum[j] = 0.f;
    }

    if (t < 32) tdm_load_rows(xbase, (void*)lbuf[0], RPC);   // wave 0 only (TDM is wave-scalar)

    for (int ch = 0; ch < NCHUNK; ++ch) {
        if (t < 32) {
            if (ch + 1 < NCHUNK) {
                tdm_load_rows(xbase + (size_t)(ch + 1) * RPC * C_, (void*)lbuf[(ch + 1) & 1], RPC);
                __builtin_amdgcn_s_wait_tensorcnt(1);   // chunk ch retired (TDM in-order per wave)
            } else {
                __builtin_amdgcn_s_wait_tensorcnt(0);
            }
        }
        __syncthreads();   // publish TDM-written LDS to all waves
        const float* buf = lbuf[ch & 1];
        float ss0 = 0.f, sd0 = 0.f, sa0 = 0.f, ss1 = 0.f, sd1 = 0.f, sa1 = 0.f;
#pragma unroll
        for (int j = 0; j < 8; ++j) {
            int c = t + 256 * j;
            float v0 = buf[c];
            float v1 = buf[C_ + c];
            ss0 += v0 * v0; sd0 += qreg[j] * v0; sa0 += v0;
            ss1 += v1 * v1; sd1 += qreg[j] * v1; sa1 += v1;
            colsum[j] += v0 + v1;
        }
#pragma unroll
        for (int m = 16; m > 0; m >>= 1) {
            ss0 += __shfl_xor(ss0, m, 32); sd0 += __shfl_xor(sd0, m, 32); sa0 += __shfl_xor(sa0, m, 32);
            ss1 += __shfl_xor(ss1, m, 32); sd1 += __shfl_xor(sd1, m, 32); sa1 += __shfl_xor(sa1, m, 32);
        }
        if (lane == 0) {
            wred[0][wave] = ss0; wred[1][wave] = sd0; wred[2][wave] = sa0;
            wred[3][wave] = ss1; wred[4][wave] = sd1; wred[5][wave] = sa1;
        }
        __syncthreads();
        if (t < 6) {
            float s = 0.f;
#pragma unroll
            for (int w = 0; w < 8; ++w) s += wred[t][w];
            int bk = b * K_ + ch * RPC + t / 3;
            int stat = t % 3;
            if (stat == 0) sq[bk] = s;
            else if (stat == 1) dot[bk] = s;
            else act[bk] = s * (1.0f / C_);
        }
        __syncthreads();   // wred + lbuf[(ch)&1] safe for reuse
    }
#pragma unroll
    for (int j = 0; j < 8; ++j)
        meanX[(size_t)b * C_ + t + 256 * j] = colsum[j] * (1.0f / K_);
}

// ---------- Row-normalize a [B,C] matrix in place; optionally emit row norms ----------
__global__ void rownorm_inplace(float* __restrict__ mat, float* __restrict__ norms) {
    __shared__ float red[256];
    int b = blockIdx.x, t = threadIdx.x;
    float* row = mat + (size_t)b * C_;
    float s = 0.f;
    for (int c = t; c < C_; c += 256) { float v = row[c]; s += v * v; }
    red[t] = s; __syncthreads();
    for (int st = 128; st > 0; st >>= 1) { if (t < st) red[t] += red[t + st]; __syncthreads(); }
    float nrm = sqrtf(red[0]);
    float inv = 1.0f / nrm;
    for (int c = t; c < C_; c += 256) row[c] *= inv;
    if (t == 0 && norms) norms[b] = nrm;
}

// ---------- NCE: l_pos = q.k, logits = [l_pos, l_neg]/T, CE target 0 ----------
__global__ void nce_loss(const float* __restrict__ q, const float* __restrict__ kmat,
                         const float* __restrict__ dot, const float* __restrict__ sq,
                         float* __restrict__ loss_out) {
    __shared__ float red[256];
    __shared__ float logits[K_ + 1];
    int b = blockIdx.x, t = threadIdx.x;
    const float* qb = q + (size_t)b * C_;
    const float* kb = kmat + (size_t)b * C_;
    float s = 0.f;
    for (int c = t; c < C_; c += 256) s += qb[c] * kb[c];
    red[t] = s; __syncthreads();
    for (int st = 128; st > 0; st >>= 1) { if (t < st) red[t] += red[t + st]; __syncthreads(); }
    if (t == 0) logits[0] = red[0] * (1.0f / T_);
    if (t < K_) logits[1 + t] = dot[b * K_ + t] * rsqrtf(sq[b * K_ + t]) * (1.0f / T_);
    __syncthreads();
    if (t == 0) {
        float m = logits[0];
        for (int i = 1; i <= K_; ++i) m = fmaxf(m, logits[i]);
        float se = 0.f;
        for (int i = 0; i <= K_; ++i) se += expf(logits[i] - m);
        loss_out[b] = (m + logf(se)) - logits[0];
    }
}

// ---------- Normalize act rows over K (one wave per row) ----------
__global__ void act_norm(const float* __restrict__ act, float* __restrict__ an) {
    int b = blockIdx.x, t = threadIdx.x;   // 32 threads
    float v = act[b * K_ + t];
    float s = v * v;
#pragma unroll
    for (int m = 16; m > 0; m >>= 1) s += __shfl_xor(s, m, 32);
    an[b * K_ + t] = v * rsqrtf(s);
}

// ---------- D = 1 - an @ an^T via V_WMMA_F32_16X16X4_F32, one wave per 16x16 tile ----------
__global__ void dmat_wmma(const float* __restrict__ an, float* __restrict__ D) {
    int I = blockIdx.x, J = blockIdx.y;    // 16 x 16 tiles
    int lane = threadIdx.x;                // 0..31, EXEC all ones
    int n = lane & 15;
    int half = lane >> 4;
    v8f c = {};
    const float* Ar = an + (size_t)(I * 16 + n) * K_;  // A frag: M = n
    const float* Br = an + (size_t)(J * 16 + n) * K_;  // B frag: B[k,n] = an[J*16+n, k]
#pragma unroll
    for (int k0 = 0; k0 < K_; k0 += 4) {
        v2f a, bf;
        a.x  = Ar[k0 + 2 * half + 0];
        a.y  = Ar[k0 + 2 * half + 1];
        bf.x = Br[k0 + 2 * half + 0];
        bf.y = Br[k0 + 2 * half + 1];
        c = __builtin_amdgcn_wmma_f32_16x16x4_f32(false, a, false, bf, (short)0, c, false, false);
    }
#pragma unroll
    for (int r = 0; r < 8; ++r) {
        D[(size_t)(I * 16 + r + 8 * half) * B_ + (J * 16 + n)] = 1.0f - c[r];
    }
}

// ---------- Triplet mining per class ----------
__global__ void triplet_cls(const float* __restrict__ label, const float* __restrict__ D,
                            float* __restrict__ trip_out) {
    __shared__ float fred[256];
    __shared__ int   ired[256];
    __shared__ int   s_anchor;
    __shared__ int   s_cnt;
    int cls = blockIdx.x, t = threadIdx.x;   // t = batch index b
    bool m = (label[t * NCLS_ + cls] == 1.0f);
    ired[t] = m ? t : -1;
    fred[t] = m ? 1.0f : 0.0f;
    __syncthreads();
    for (int st = 128; st > 0; st >>= 1) {
        if (t < st) { ired[t] = max(ired[t], ired[t + st]); fred[t] += fred[t + st]; }
        __syncthreads();
    }
    if (t == 0) { s_anchor = ired[0]; s_cnt = (int)fred[0]; }
    __syncthreads();
    int anchor = s_anchor;
    float row = (anchor >= 0) ? D[(size_t)anchor * B_ + t] : 0.0f;
    const float INF = __builtin_inff();
    fred[t] = (m && t != anchor) ? row : -INF;
    __syncthreads();
    for (int st = 128; st > 0; st >>= 1) { if (t < st) fred[t] = fmaxf(fred[t], fred[t + st]); __syncthreads(); }
    float max_mo = fred[0];
    __syncthreads();
    fred[t] = row;
    __syncthreads();
    for (int st = 128; st > 0; st >>= 1) { if (t < st) fred[t] = fmaxf(fred[t], fred[t + st]); __syncthreads(); }
    float max_all = fred[0];
    __syncthreads();
    fred[t] = (!m) ? row : INF;
    __syncthreads();
    for (int st = 128; st > 0; st >>= 1) { if (t < st) fred[t] = fminf(fred[t], fred[t + st]); __syncthreads(); }
    float min_nm = fred[0];
    if (t == 0) {
        float max_d = fmaxf(0.0f, max_mo);
        float min_d = fminf(max_all, min_nm);
        float contrib = fmaxf(max_d - min_d + TRIP_MARGIN_, 0.0f);
        trip_out[cls] = (s_cnt > 1) ? contrib : 0.0f;
    }
}

// ---------- BCE with row-normalized labels ----------
__global__ void bce_loss(const float* __restrict__ scores, const float* __restrict__ label,
                         float* __restrict__ out_v) {
    __shared__ float red[256];
    int t = threadIdx.x;   // t = batch index
    const float* lb = label + t * NCLS_;
    const float* p  = scores + t * NCLS_;
    float rs = 0.f;
#pragma unroll
    for (int j = 0; j < NCLS_; ++j) rs += lb[j];
    float inv = 1.0f / rs;
    float acc = 0.f;
#pragma unroll
    for (int j = 0; j < NCLS_; ++j) {
        float l = lb[j] * inv;
        acc += -(l * logf(p[j]) + (1.0f - l) * logf(1.0f - p[j]));
    }
    red[t] = acc; __syncthreads();
    for (int st = 128; st > 0; st >>= 1) { if (t < st) red[t] += red[t + st]; __syncthreads(); }
    if (t == 0) out_v[0] = red[0] / (float)(B_ * NCLS_);
}

// ---------- Abs (uncertainty-margin) loss ----------
__global__ void abs_loss(const float* __restrict__ normPA, const float* __restrict__ normPB,
                         float* __restrict__ out_v) {
    __shared__ float red[256];
    int t = threadIdx.x;
    float la = fmaxf(MARGIN_ - normPA[t], 0.0f);
    float v = la + normPB[t];
    red[t] = v * v; __syncthreads();
    for (int st = 128; st > 0; st >>= 1) { if (t < st) red[t] += red[t + st]; __syncthreads(); }
    if (t == 0) out_v[0] = red[0] / (float)B_;
}

// ---------- Combine to the 5 outputs ----------
__global__ void combine(const float* __restrict__ n1, const float* __restrict__ n2,
                        const float* __restrict__ trip, const float* __restrict__ bce_v,
                        const float* __restrict__ abs_v, float* __restrict__ out) {
    __shared__ float r1[256], r2[256];
    int t = threadIdx.x;
    r1[t] = n1[t]; r2[t] = n2[t];
    __syncthreads();
    for (int st = 128; st > 0; st >>= 1) { if (t < st) { r1[t] += r1[t + st]; r2[t] += r2[t + st]; } __syncthreads(); }
    if (t == 0) {
        float snico = r1[0] / (float)B_ + r2[0] / (float)B_;
        float ts = 0.f;
        for (int i = 0; i < NCLS_; ++i) ts += trip[i];
        float cls = bce_v[0], ab = abs_v[0];
        float total = cls + 0.01f * snico + 0.0005f * ab + 0.005f * ts;
        out[0] = total; out[1] = cls; out[2] = snico; out[3] = ab; out[4] = ts;
    }
}

extern "C" void kernel_launch(void* const* d_in, const int* in_sizes, int n_in,
                              void* d_out, int out_size, void* d_ws, size_t ws_size,
                              hipStream_t stream) {
    (void)in_sizes; (void)n_in; (void)out_size; (void)ws_size;
    const float* scores = (const float*)d_in[0];
    const float* label  = (const float*)d_in[1];
    const float* PA     = (const float*)d_in[2];
    const float* PB     = (const float*)d_in[3];
    const float* NA_    = (const float*)d_in[4];
    const float* NB_    = (const float*)d_in[5];
    float* out = (float*)d_out;

    const size_t BC = (size_t)B_ * C_;   // 524288
    const size_t BK = (size_t)B_ * K_;   // 8192
    float* w = (float*)d_ws;
    float* meanNA = w;              // becomes q1
    float* meanNB = w + BC;         // becomes q2
    float* meanPA = w + 2 * BC;     // becomes kA
    float* meanPB = w + 3 * BC;     // becomes kB
    float* sqPA   = w + 4 * BC;
    float* dotPA  = sqPA + BK;
    float* actPA  = dotPA + BK;
    float* sqPB   = actPA + BK;
    float* dotPB  = sqPB + BK;
    float* actPB  = dotPB + BK;
    float* normPA = actPB + BK;
    float* normPB = normPA + B_;
    float* an     = normPB + B_;
    float* lossN1 = an + BK;
    float* lossN2 = lossN1 + B_;
    float* tripv  = lossN2 + B_;
    float* bcev   = tripv + NCLS_;
    float* absv   = bcev + 1;
    float* Dm     = absv + 1;       // 256*256 floats

    const int nb_mean = (int)(BC / 256);
    mean_over_k<<<nb_mean, 256, 0, stream>>>(NA_, meanNA);
    mean_over_k<<<nb_mean, 256, 0, stream>>>(NB_, meanNB);

    rownorm_inplace<<<B_, 256, 0, stream>>>(meanNA, nullptr);   // q1
    rownorm_inplace<<<B_, 256, 0, stream>>>(meanNB, nullptr);   // q2

    // single TDM-staged read of PA / PB produces mean-over-K, col-norms^2, q-dots, act
    fused_stats<<<B_, 256, 0, stream>>>(PA, meanNB /*q2*/, meanPA, sqPA, dotPA, actPA);
    fused_stats<<<B_, 256, 0, stream>>>(PB, meanNA /*q1*/, meanPB, sqPB, dotPB, actPB);

    rownorm_inplace<<<B_, 256, 0, stream>>>(meanPA, normPA);    // kA + ||PA.mean||
    rownorm_inplace<<<B_, 256, 0, stream>>>(meanPB, normPB);    // kB + ||PB.mean||

    nce_loss<<<B_, 256, 0, stream>>>(meanNA /*q1*/, meanPA /*kA*/, dotPB, sqPB, lossN1);
    nce_loss<<<B_, 256, 0, stream>>>(meanNB /*q2*/, meanPB /*kB*/, dotPA, sqPA, lossN2);

    act_norm<<<B_, 32, 0, stream>>>(actPA, an);
    dmat_wmma<<<dim3(16, 16), 32, 0, stream>>>(an, Dm);
    triplet_cls<<<NCLS_, 256, 0, stream>>>(label, Dm, tripv);

    bce_loss<<<1, 256, 0, stream>>>(scores, label, bcev);
    abs_loss<<<1, 256, 0, stream>>>(normPA, normPB, absv);
    combine<<<1, 256, 0, stream>>>(lossN1, lossN2, tripv, bcev, absv, out);
}